// BlockWavelet_Down_71347996721578
// MI455X (gfx1250) — compile-verified
//
#include <hip/hip_runtime.h>
#include <math.h>

// BlockWavelet_Down: fused depthwise conv7 -> BN -> GELU(exact) -> leaky ->
// Haar split -> depthwise conv7/conv3 -> Haar merge -> 1x1 skip -> leaky.
// One block per (batch, in-channel) row; x row staged into LDS via CDNA5
// async global->LDS copies (global_load_async_to_lds_b128 / s_wait_asynccnt).

constexpr int B    = 64;
constexpr int CIN  = 64;
constexpr int COUT = 128;
constexpr int L    = 4096;
constexpr int LH   = 2048;
constexpr float INV_SQRT2 = 0.70710678118654752440f;
constexpr float EPSF = 1e-5f;

__device__ __forceinline__ unsigned lds_byte_off(const void* p) {
    // Generic pointer to LDS: low 32 bits are the wave-relative LDS byte offset.
    return (unsigned)(unsigned long long)p;
}

__global__ __launch_bounds__(256, 4)
void wavelet_down_kernel(const float* __restrict__ x,
                         const float* __restrict__ w1,
                         const float* __restrict__ g1,
                         const float* __restrict__ b1,
                         const float* __restrict__ m1,
                         const float* __restrict__ v1,
                         const float* __restrict__ w2L,
                         const float* __restrict__ w2H,
                         const float* __restrict__ wsk,
                         float* __restrict__ out)
{
    // LDS carving (floats):
    //   xs : 4104  (x row at offset 4; left pad 4 [need 3], right pad 4 [need 3])
    //   hs : 4096  (post conv1+bn+gelu+leaky)
    //   xl : 2054  (interior at +3, 3-wide zero halos)
    //   xh : 2050  (interior at +1, 1-wide zero halos)
    __shared__ __align__(16) float smem[12304];   // 49216 bytes
    float* xs = smem;
    float* hs = smem + 4104;
    float* xl = smem + 8200;
    float* xh = smem + 10254;

    const int tid = threadIdx.x;
    const int blk = blockIdx.x;
    const int b   = blk >> 6;    // / CIN
    const int c   = blk & 63;    // % CIN

    // Zero the halos (interior is overwritten every phase; halos stay zero).
    if (tid < 4) { xs[tid] = 0.0f; xs[4100 + tid] = 0.0f; }
    if (tid < 3) { xl[tid] = 0.0f; xl[2051 + tid] = 0.0f; }
    if (tid == 0) { xh[0] = 0.0f; xh[2049] = 0.0f; }

    // ---- Async copy of the 16KB x row into LDS (CDNA5 async-to-LDS path) ----
    const float* gsrc = x + ((size_t)b * CIN + c) * L;
    const unsigned lbase = lds_byte_off(&xs[4]);   // 16B aligned (offset 16 from smem)
#pragma unroll
    for (int j = 0; j < 4; ++j) {
        const int e = (j * 256 + tid) * 4;         // float index, 16B granularity
        const unsigned ldst = lbase + (unsigned)e * 4u;
        const float* gaddr = gsrc + e;
        asm volatile("global_load_async_to_lds_b128 %0, %1, off"
                     :: "v"(ldst), "v"(gaddr) : "memory");
    }
    asm volatile("s_wait_asynccnt 0" ::: "memory");
    __syncthreads();

    // ---- Two output channels per input channel (groups=CIN, Cout/groups=2) ----
    for (int sub = 0; sub < 2; ++sub) {
        const int oc = 2 * c + sub;

        float k1[7];
#pragma unroll
        for (int k = 0; k < 7; ++k) k1[k] = w1[oc * 7 + k];
        const float inv = g1[oc] / sqrtf(v1[oc] + EPSF);
        const float bia = b1[oc] - m1[oc] * inv;

        // conv7 (pad 3) + BN + exact GELU + leaky -> hs
        for (int t = tid; t < L; t += 256) {
            float a = 0.0f;
#pragma unroll
            for (int k = 0; k < 7; ++k) a = fmaf(xs[t + 1 + k], k1[k], a);
            a = fmaf(a, inv, bia);
            a = 0.5f * a * (1.0f + erff(a * INV_SQRT2));
            a = (a >= 0.0f) ? a : 0.01f * a;
            hs[t] = a;
        }
        __syncthreads();

        // Haar lifting: even/odd -> xl, xh
        for (int i = tid; i < LH; i += 256) {
            const float e = hs[2 * i];
            const float o = hs[2 * i + 1];
            xl[3 + i] = (e + o) * INV_SQRT2;
            xh[1 + i] = (e - o) * INV_SQRT2;
        }
        __syncthreads();

        float kL[7], kH[3];
#pragma unroll
        for (int k = 0; k < 7; ++k) kL[k] = w2L[oc * 7 + k];
#pragma unroll
        for (int k = 0; k < 3; ++k) kH[k] = w2H[oc * 3 + k];
        const float ws = wsk[oc];

        float2* __restrict__ orow =
            (float2*)(out + ((size_t)b * COUT + oc) * L);

        // conv7(xl) / conv3(xh) -> reconstruct -> +skip -> leaky -> store
        for (int i = tid; i < LH; i += 256) {
            float aL = 0.0f, aH = 0.0f;
#pragma unroll
            for (int k = 0; k < 7; ++k) aL = fmaf(xl[i + k], kL[k], aL);
#pragma unroll
            for (int k = 0; k < 3; ++k) aH = fmaf(xh[i + k], kH[k], aH);
            const float re = (aL + aH) * INV_SQRT2;
            const float ro = (aL - aH) * INV_SQRT2;
            float o0 = fmaf(xs[4 + 2 * i],     ws, re);
            float o1 = fmaf(xs[4 + 2 * i + 1], ws, ro);
            o0 = (o0 >= 0.0f) ? o0 : 0.01f * o0;
            o1 = (o1 >= 0.0f) ? o1 : 0.01f * o1;
            orow[i] = make_float2(o0, o1);
        }
        __syncthreads();   // hs/xl/xh reused by next sub-channel
    }
}

extern "C" void kernel_launch(void* const* d_in, const int* in_sizes, int n_in,
                              void* d_out, int out_size, void* d_ws, size_t ws_size,
                              hipStream_t stream)
{
    (void)in_sizes; (void)n_in; (void)d_ws; (void)ws_size; (void)out_size;
    const float* x   = (const float*)d_in[0];
    const float* w1  = (const float*)d_in[1];
    const float* g1  = (const float*)d_in[2];
    const float* b1  = (const float*)d_in[3];
    const float* m1  = (const float*)d_in[4];
    const float* v1  = (const float*)d_in[5];
    const float* w2L = (const float*)d_in[6];
    const float* w2H = (const float*)d_in[7];
    const float* wsk = (const float*)d_in[8];
    float* out = (float*)d_out;

    dim3 grid(B * CIN);   // 4096 blocks, one per (batch, in-channel) row
    dim3 block(256);      // 8 wave32 per block
    wavelet_down_kernel<<<grid, block, 0, stream>>>(
        x, w1, g1, b1, m1, v1, w2L, w2H, wsk, out);
}